// MultiHeadAttentionWithGeo_50225347559757
// MI455X (gfx1250) — compile-verified
//
#include <hip/hip_runtime.h>

// Problem constants (from reference): B=2, NQ=NK=512, D=256, H=8, dh=32
#define B_   2
#define NQ_  512
#define NK_  512
#define D_   256
#define H_   8
#define DH_  32

typedef __attribute__((ext_vector_type(2))) float v2f;   // A/B frag for f32 16x16x4
typedef __attribute__((ext_vector_type(8))) float v8f;   // C/D frag (16x16 f32)

// D = A(16x4) * B(4x16) + C(16x16), all fp32, wave32.
__device__ __forceinline__ v8f wmma_f32_4(v2f a, v2f b, v8f c) {
  return __builtin_amdgcn_wmma_f32_16x16x4_f32(
      /*neg_a=*/false, a, /*neg_b=*/false, b,
      /*c_mod=*/(short)0, c, /*reuse_a=*/false, /*reuse_b=*/false);
}

// ---- CDNA5 async global->LDS staging (ASYNCcnt-tracked), with safe fallback ----
#if defined(__gfx1250__) && __has_builtin(__builtin_amdgcn_global_load_async_to_lds_b128)
#define HAVE_ASYNC_LDS 1
#else
#define HAVE_ASYNC_LDS 0
#endif

#if HAVE_ASYNC_LDS
typedef int v4i_ __attribute__((vector_size(4 * sizeof(int))));
typedef __attribute__((address_space(1))) v4i_* gv4i_p;  // global (__device__) AS
typedef __attribute__((address_space(3))) v4i_* lv4i_p;  // LDS (__shared__) AS

__device__ __forceinline__ void async_copy_b128(const float* g, float* l) {
  __builtin_amdgcn_global_load_async_to_lds_b128((gv4i_p)(g), (lv4i_p)(l), 0, 0);
}
#if __has_builtin(__builtin_amdgcn_s_wait_asynccnt)
#define ASYNC_WAIT(n) __builtin_amdgcn_s_wait_asynccnt(n)
#else
#define ASYNC_WAIT(n) asm volatile("s_wait_asynccnt %0" ::"n"(n))
#endif
#endif

// -----------------------------------------------------------------------------
// Projection GEMM: Out[n][j] = relu( sum_d X[n][d] * W[j][d] + bias[j] )
// X: [1024, 256] (B*N rows), W: [256, 256] row-major.
// One wave computes a 16x16 output tile; 4 waves/block cover 4 col-tiles.
// HEAD_SPLIT: write [B][H][N][32] layout; else plain [B*N][256].
// -----------------------------------------------------------------------------
template <bool HEAD_SPLIT>
__global__ __launch_bounds__(128) void proj_kernel(
    const float* __restrict__ X, const float* __restrict__ W,
    const float* __restrict__ bias, float* __restrict__ out) {
  const int lane = threadIdx.x & 31;
  const int wave = threadIdx.x >> 5;
  const int m    = lane & 15;
  const int hi   = lane >> 4;              // lane half: selects K pair
  const int nt   = blockIdx.x;             // row tile (0..63)
  const int jt   = blockIdx.y * 4 + wave;  // col tile (0..15)

  const float* xrow = X + (size_t)(nt * 16 + m) * D_;  // A: row = M
  const float* wrow = W + (size_t)(jt * 16 + m) * D_;  // B: lane = N, reads W[n][d]

  v8f acc = {};
  for (int d0 = 0; d0 < D_; d0 += 4) {
    v2f a, b;
    const int dk = d0 + hi * 2;
    a.x = xrow[dk];     a.y = xrow[dk + 1];
    b.x = wrow[dk];     b.y = wrow[dk + 1];
    acc = wmma_f32_4(a, b, acc);
  }

  const int j = jt * 16 + m;  // output column (C/D: N = lane&15)
  const float bj = bias[j];
#pragma unroll
  for (int i = 0; i < 8; ++i) {
    const int mm = i + hi * 8;        // C/D: M = i (+8 for upper lanes)
    const int n  = nt * 16 + mm;      // global row in [0, B*N)
    float val = fmaxf(acc[i] + bj, 0.0f);
    if (HEAD_SPLIT) {
      const int bb = n / NQ_, nn = n % NQ_;
      const int h = j >> 5, jd = j & 31;
      out[(((size_t)(bb * H_ + h)) * NQ_ + nn) * DH_ + jd] = val;
    } else {
      out[(size_t)n * D_ + j] = val;
    }
  }
}

// -----------------------------------------------------------------------------
// Geo bias: glog[b][h][q][k] = log( relu( sum_d pos[b][d][q][k]*Wg[h][d] + bg[h] ) + 1e-6 )
// HBM-bound: streams pos_embedding (512 MB) exactly once.
// WMMA tile: M=16 (8 heads + 8 zero pad), N=16 k-cols, K-loop over d (256).
// One block per (b,q). Async path: double-buffered GLOBAL_LOAD_ASYNC_TO_LDS_B128
// stages 8-d-row x 512-k chunks (16 KB) so DMA of chunk c+1 overlaps WMMA on c.
// 8 waves; wave w owns k-columns [32w,32w+32) and [32(w+8),32(w+8)+32).
// -----------------------------------------------------------------------------
#define DCHUNK 8
#define NCHUNK (D_ / DCHUNK)  // 32

__global__ __launch_bounds__(256) void geo_kernel(
    const float* __restrict__ pos, const float* __restrict__ Wg,
    const float* __restrict__ bg, float* __restrict__ glog) {
  const int lane = threadIdx.x & 31;
  const int wave = threadIdx.x >> 5;  // 0..7
  const int m    = lane & 15;
  const int hi   = lane >> 4;
  const int bq = blockIdx.x;
  const int b = bq / NQ_, q = bq % NQ_;
  const int mh = m & 7;  // clamp for padded A rows (avoid OOB reads)

  const int kb0 = wave * 32;        // task-0 k-base
  const int kb1 = (wave + 8) * 32;  // task-1 k-base

#if HAVE_ASYNC_LDS
  __shared__ float buf[2][DCHUNK * NK_];  // 2 x 16 KB double buffer
  const int tid = threadIdx.x;
  const float* gbase = pos + ((size_t)(b * D_) * NQ_ + q) * NK_;

  // Issue one chunk's async DMA: 8 rows x 512 floats = 1024 float4; 4 per thread.
  auto issue_chunk = [&](int c) {
    const float* gsrc = gbase + (size_t)(c * DCHUNK) * (NQ_ * NK_);
    float* lbase = &buf[c & 1][0];
#pragma unroll
    for (int jj = 0; jj < 4; ++jj) {
      const int idx = jj * 256 + tid;  // float4 index within chunk
      const int dl  = idx >> 7;        // 0..7 (128 float4 per 512-float row)
      const int kv  = (idx & 127) * 4;
      async_copy_b128(gsrc + (size_t)dl * (NQ_ * NK_) + kv, lbase + dl * NK_ + kv);
    }
  };

  v8f a00 = {}, a01 = {}, a10 = {}, a11 = {};  // [task][k-subtile] accumulators
  issue_chunk(0);
#pragma unroll 1  // keep accumulators in place: no unroll-by-2 register rotation
  for (int c = 0; c < NCHUNK; ++c) {
    if (c + 1 < NCHUNK) {
      issue_chunk(c + 1);
      ASYNC_WAIT(4);  // in-order: chunk c's 4 loads complete, c+1 may be in flight
    } else {
      ASYNC_WAIT(0);
    }
    __syncthreads();  // all waves' slice of chunk c visible in LDS

    const float* lb = &buf[c & 1][0];
    const int dbase = c * DCHUNK;

    // A fragments (Wg rows, heads 0..7, zero-pad 8..15) for this chunk.
    float ax[2], ay[2];
#pragma unroll
    for (int i = 0; i < 2; ++i) {
      const int dkg = dbase + i * 4 + hi * 2;
      const float w0 = Wg[mh * D_ + dkg];
      const float w1 = Wg[mh * D_ + dkg + 1];
      ax[i] = (m < H_) ? w0 : 0.0f;
      ay[i] = (m < H_) ? w1 : 0.0f;
    }
#pragma unroll
    for (int i = 0; i < 2; ++i) {
      const int dl = i * 4 + hi * 2;  // local d row pair for this lane half
      v2f a;  a.x = ax[i];  a.y = ay[i];
      v2f b00, b01, b10, b11;
      b00.x = lb[dl * NK_ + kb0 + m];       b00.y = lb[(dl + 1) * NK_ + kb0 + m];
      b01.x = lb[dl * NK_ + kb0 + 16 + m];  b01.y = lb[(dl + 1) * NK_ + kb0 + 16 + m];
      b10.x = lb[dl * NK_ + kb1 + m];       b10.y = lb[(dl + 1) * NK_ + kb1 + m];
      b11.x = lb[dl * NK_ + kb1 + 16 + m];  b11.y = lb[(dl + 1) * NK_ + kb1 + 16 + m];
      a00 = wmma_f32_4(a, b00, a00);
      a01 = wmma_f32_4(a, b01, a01);
      a10 = wmma_f32_4(a, b10, a10);
      a11 = wmma_f32_4(a, b11, a11);
    }
    __syncthreads();  // buffer (c&1) free for reuse at iteration c+1's issue
  }

  // D layout: lanes 0-15, VGPR i -> head h=i (i<8); upper lanes = padding.
  if (hi == 0) {
#pragma unroll
    for (int i = 0; i < 8; ++i) {
      const float bgi = bg[i];
      const size_t base = (((size_t)(b * H_ + i)) * NQ_ + q) * NK_;
      glog[base + kb0 + m]      = __logf(fmaxf(a00[i] + bgi, 0.0f) + 1e-6f);
      glog[base + kb0 + 16 + m] = __logf(fmaxf(a01[i] + bgi, 0.0f) + 1e-6f);
      glog[base + kb1 + m]      = __logf(fmaxf(a10[i] + bgi, 0.0f) + 1e-6f);
      glog[base + kb1 + 16 + m] = __logf(fmaxf(a11[i] + bgi, 0.0f) + 1e-6f);
    }
  }
#else
  // Fallback: direct global->VGPR B fragments.
  for (int task = wave; task < NK_ / 32; task += 8) {
    const int kbase = task * 32;
    v8f c0 = {}, c1 = {};
    for (int d0 = 0; d0 < D_; d0 += 4) {
      const int dk = d0 + hi * 2;
      v2f a, b0v, b1v;
      const float w0 = Wg[mh * D_ + dk];
      const float w1 = Wg[mh * D_ + dk + 1];
      a.x = (m < H_) ? w0 : 0.0f;
      a.y = (m < H_) ? w1 : 0.0f;
      const float* p0 = pos + (((size_t)(b * D_ + dk)) * NQ_ + q) * NK_;
      const float* p1 = pos + (((size_t)(b * D_ + dk + 1)) * NQ_ + q) * NK_;
      b0v.x = p0[kbase + m];        b0v.y = p1[kbase + m];
      b1v.x = p0[kbase + 16 + m];   b1v.y = p1[kbase + 16 + m];
      c0 = wmma_f32_4(a, b0v, c0);
      c1 = wmma_f32_4(a, b1v, c1);
    }
    if (hi == 0) {
#pragma unroll
      for (int i = 0; i < 8; ++i) {
        const float g0 = fmaxf(c0[i] + bg[i], 0.0f);
        const float g1 = fmaxf(c1[i] + bg[i], 0.0f);
        const size_t base = (((size_t)(b * H_ + i)) * NQ_ + q) * NK_;
        glog[base + kbase + m]      = __logf(g0 + 1e-6f);
        glog[base + kbase + 16 + m] = __logf(g1 + 1e-6f);
      }
    }
  }
#endif
}

// -----------------------------------------------------------------------------
// Attention core per (b, h, 16-q-tile): scores = glog + QK^T/sqrt(32), mask,
// softmax over NK, then Y = P @ V. 2 waves per block.
// -----------------------------------------------------------------------------
#define SC_PITCH 516  // 512 + 4 pad: LDS bank-conflict-free A-frag reads
__global__ __launch_bounds__(64) void attn_kernel(
    const float* __restrict__ Qh, const float* __restrict__ Kh,
    const float* __restrict__ Vh, const float* __restrict__ glog,
    const int* __restrict__ mask, float* __restrict__ Y) {
  __shared__ float sc[16][SC_PITCH];
  __shared__ float red[16][4];
  __shared__ float rowm[16];
  __shared__ float rowinv[16];

  const int lane = threadIdx.x & 31;
  const int wave = threadIdx.x >> 5;  // 0..1
  const int m    = lane & 15;
  const int hi   = lane >> 4;

  const int qt = blockIdx.x;  // 0..31
  const int h  = blockIdx.y;  // 0..7
  const int b  = blockIdx.z;  // 0..1
  const int q0 = qt * 16;

  const float* Qb = Qh + ((size_t)(b * H_ + h)) * NQ_ * DH_;
  const float* Kb = Kh + ((size_t)(b * H_ + h)) * NK_ * DH_;
  const float* Vb = Vh + ((size_t)(b * H_ + h)) * NK_ * DH_;
  const float* Gb = glog + ((size_t)(b * H_ + h)) * NQ_ * NK_;
  const int*   Mb = mask + ((size_t)b) * NQ_ * NK_;

  // Preload Q A-fragments for the whole dh=32 reduction (8 iters of K=4).
  float qax[8], qay[8];
#pragma unroll
  for (int it = 0; it < 8; ++it) {
    const int dk = it * 4 + hi * 2;
    qax[it] = Qb[(size_t)(q0 + m) * DH_ + dk];
    qay[it] = Qb[(size_t)(q0 + m) * DH_ + dk + 1];
  }

  const float scale = 0.17677669529663687f;  // 1/sqrt(32)

  // ---- scores: each wave covers alternating 16-wide k tiles ----
#pragma unroll 1
  for (int t = wave; t < NK_ / 16; t += 2) {
    const int k0 = t * 16;
    v8f acc = {};
#pragma unroll
    for (int it = 0; it < 8; ++it) {
      const int dk = it * 4 + hi * 2;
      v2f a, bf;
      a.x = qax[it];  a.y = qay[it];
      bf.x = Kb[(size_t)(k0 + m) * DH_ + dk];      // B: lane = N (k col), rows = d
      bf.y = Kb[(size_t)(k0 + m) * DH_ + dk + 1];
      acc = wmma_f32_4(a, bf, acc);
    }
#pragma unroll
    for (int i = 0; i < 8; ++i) {
      const int mm = i + hi * 8;
      const int q  = q0 + mm;
      const int k  = k0 + m;
      float s = acc[i] * scale + Gb[(size_t)q * NK_ + k];
      if (Mb[(size_t)q * NK_ + k] == 0) s = -1e9f;
      sc[mm][k] = s;
    }
  }
  __syncthreads();

  // ---- softmax over 512 cols: 16 rows x 4 threads x 128 cols ----
  {
    const int r  = threadIdx.x >> 2;
    const int c4 = threadIdx.x & 3;
    float lmax = -3.4e38f;
    for (int c = c4 * 128; c < c4 * 128 + 128; ++c) lmax = fmaxf(lmax, sc[r][c]);
    red[r][c4] = lmax;
    __syncthreads();
    if (threadIdx.x < 16) {
      const int rr = threadIdx.x;
      rowm[rr] = fmaxf(fmaxf(red[rr][0], red[rr][1]), fmaxf(red[rr][2], red[rr][3]));
    }
    __syncthreads();
    const float rm = rowm[r];
    float lsum = 0.0f;
    for (int c = c4 * 128; c < c4 * 128 + 128; ++c) {
      const float e = __expf(sc[r][c] - rm);
      sc[r][c] = e;
      lsum += e;
    }
    red[r][c4] = lsum;
    __syncthreads();
    if (threadIdx.x < 16) {
      const int rr = threadIdx.x;
      rowinv[rr] = 1.0f / (red[rr][0] + red[rr][1] + red[rr][2] + red[rr][3]);
    }
    __syncthreads();
  }

  // ---- Y tile = P(16x512) @ V(512x16): wave w owns d-cols [16w, 16w+16) ----
  {
    const int dt = wave * 16;
    v8f acc = {};
#pragma unroll 1
    for (int k0 = 0; k0 < NK_; k0 += 4) {
      const int kk = k0 + hi * 2;
      v2f a, bf;
      a.x = sc[m][kk];            // A frag from LDS (P tile, padded pitch)
      a.y = sc[m][kk + 1];
      bf.x = Vb[(size_t)kk * DH_ + dt + m];
      bf.y = Vb[(size_t)(kk + 1) * DH_ + dt + m];
      acc = wmma_f32_4(a, bf, acc);
    }
#pragma unroll
    for (int i = 0; i < 8; ++i) {
      const int mm = i + hi * 8;
      const int q  = q0 + mm;
      Y[((size_t)(b * NQ_) + q) * D_ + h * DH_ + dt + m] = acc[i] * rowinv[mm];
    }
  }
}

// -----------------------------------------------------------------------------
extern "C" void kernel_launch(void* const* d_in, const int* in_sizes, int n_in,
                              void* d_out, int out_size, void* d_ws, size_t ws_size,
                              hipStream_t stream) {
  const float* q    = (const float*)d_in[0];
  const float* k    = (const float*)d_in[1];
  const float* v    = (const float*)d_in[2];
  const float* pos  = (const float*)d_in[3];
  const int*   mask = (const int*)d_in[4];
  const float* Wq   = (const float*)d_in[5];
  const float* bq   = (const float*)d_in[6];
  const float* Wk   = (const float*)d_in[7];
  const float* bk   = (const float*)d_in[8];
  const float* Wv   = (const float*)d_in[9];
  const float* bv   = (const float*)d_in[10];
  const float* Wo   = (const float*)d_in[11];
  const float* bo   = (const float*)d_in[12];
  const float* Wg   = (const float*)d_in[13];
  const float* bg   = (const float*)d_in[14];

  float* ws = (float*)d_ws;
  float* Qh = ws;                                   // [B][H][NQ][32]  1 MB
  float* Kh = Qh + (size_t)B_ * H_ * NQ_ * DH_;     // [B][H][NK][32]  1 MB
  float* Vh = Kh + (size_t)B_ * H_ * NK_ * DH_;     // [B][H][NK][32]  1 MB
  float* Yb = Vh + (size_t)B_ * H_ * NK_ * DH_;     // [B][NQ][256]    1 MB
  float* gl = Yb + (size_t)B_ * NQ_ * D_;           // [B][H][NQ][NK] 16 MB

  const dim3 pgrid(64, 4), pblk(128);
  proj_kernel<true><<<pgrid, pblk, 0, stream>>>(q, Wq, bq, Qh);
  proj_kernel<true><<<pgrid, pblk, 0, stream>>>(k, Wk, bk, Kh);
  proj_kernel<true><<<pgrid, pblk, 0, stream>>>(v, Wv, bv, Vh);
  geo_kernel<<<dim3(B_ * NQ_), 256, 0, stream>>>(pos, Wg, bg, gl);
  attn_kernel<<<dim3(NQ_ / 16, H_, B_), 64, 0, stream>>>(Qh, Kh, Vh, gl, mask, Yb);
  proj_kernel<false><<<pgrid, pblk, 0, stream>>>(Yb, Wo, bo, (float*)d_out);
}